// ReUploadingPQC_8031588843540
// MI455X (gfx1250) — compile-verified
//
#include <hip/hip_runtime.h>
#include <hip/hip_bf16.h>
#include <math.h>

#define N_QUBITS 4
#define N_LAYERS 8

typedef __attribute__((ext_vector_type(2))) float v2f;
typedef __attribute__((ext_vector_type(8))) float v8f;

// d_ws float layout:
//  [0 .. 6912)  : U matrices packed in WMMA A-operand layout.
//                 idxA(l,v,c) = ((l*3+v)*4+c)*64 ; element at + lane*2 + p
//                 holds M_v[m = lane&15, k = 4c + p + 2*(lane>>4)]
//                 v: 0 = Re(U_l), 1 = Im(U_l), 2 = -Im(U_l)
//  [6912..6928) : state0 = U_0|0...0>  real part (16 amps)
//  [6928..6944) : state0 imag part (16 amps)

struct c2f { float re, im; };
__device__ inline c2f cmul(c2f a, c2f b) { return { a.re*b.re - a.im*b.im, a.re*b.im + a.im*b.re }; }
__device__ inline c2f cadd(c2f a, c2f b) { return { a.re + b.re, a.im + b.im }; }
__device__ inline void mm2(const c2f A[2][2], const c2f B[2][2], c2f C[2][2]) {
  for (int i = 0; i < 2; ++i)
    for (int j = 0; j < 2; ++j)
      C[i][j] = cadd(cmul(A[i][0], B[0][j]), cmul(A[i][1], B[1][j]));
}

// One block per layer matrix l = 0..8; thread t -> entry (m = t>>4, k = t&15).
__global__ void pqc_setup(const float* __restrict__ theta, float* __restrict__ ws)
{
  const int l = blockIdx.x;        // 0..8
  const int t = threadIdx.x;       // 0..255
  const int m = t >> 4;            // output amplitude index (row)
  const int k = t & 15;            // input amplitude index (col)

  // Combined single-qubit gates G_q = RZ * RY * RX for this layer.
  c2f G[N_QUBITS][2][2];
  for (int q = 0; q < N_QUBITS; ++q) {
    const float ax = theta[(l*N_QUBITS + q)*3 + 0];
    const float ay = theta[(l*N_QUBITS + q)*3 + 1];
    const float az = theta[(l*N_QUBITS + q)*3 + 2];
    const float cx = cosf(0.5f*ax), sx = sinf(0.5f*ax);
    const float cy = cosf(0.5f*ay), sy = sinf(0.5f*ay);
    const float cz = cosf(0.5f*az), sz = sinf(0.5f*az);
    const c2f RX[2][2] = { { {cx,0.f}, {0.f,-sx} }, { {0.f,-sx}, {cx,0.f} } };
    const c2f RY[2][2] = { { {cy,0.f}, {-sy,0.f} }, { {sy,0.f}, {cy,0.f} } };
    const c2f RZ[2][2] = { { {cz,-sz}, {0.f,0.f} }, { {0.f,0.f}, {cz,sz} } };
    c2f T[2][2];
    mm2(RY, RX, T);
    mm2(RZ, T, G[q]);
  }

  // U[m][k] = prod_q G_q[bit_q(m)][bit_q(k)]; qubit q maps to amp-index bit (3-q).
  c2f val = {1.f, 0.f};
  for (int q = 0; q < N_QUBITS; ++q) {
    const int bm = (m >> (3 - q)) & 1;
    const int bk = (k >> (3 - q)) & 1;
    val = cmul(val, G[q][bm][bk]);
  }
  if (l < N_LAYERS) {              // ring-CZ diagonal (applied after rotations)
    float sgn = 1.f;
    for (int i = 0; i < N_QUBITS; ++i) {
      const int j  = (i + 1) & 3;
      const int bi = (m >> (3 - i)) & 1;
      const int bj = (m >> (3 - j)) & 1;
      if (bi & bj) sgn = -sgn;
    }
    val.re *= sgn; val.im *= sgn;
  }

  // Scatter into WMMA A-operand layout: k -> (chunk c, vgpr-slot p, lane-half hi).
  const int c    = k >> 2;
  const int p    = k & 1;
  const int hi   = (k >> 1) & 1;
  const int lane = m + 16*hi;
  const int base = ((l*3 + 0)*4 + c)*64 + lane*2 + p;
  ws[base]         =  val.re;      // Re(U)
  ws[base + 256]   =  val.im;      // Im(U)
  ws[base + 512]   = -val.im;      // -Im(U)

  if (l == 0 && k == 0) {          // |psi0> = U_0 column 0
    ws[6912 + m] = val.re;
    ws[6928 + m] = val.im;
  }
}

// One wave32 per 16 samples. State kept in WMMA C/D layout:
// VGPR r holds amplitude (r + 8*(lane>=16)) of sample (tile*16 + (lane&15)).
__global__ void pqc_main(const float* __restrict__ x,
                         const float* __restrict__ lmbd,
                         const float* __restrict__ ws,
                         float* __restrict__ out, int Bn)
{
  const int lane = threadIdx.x & 31;
  const int wave = (int)((blockIdx.x * blockDim.x + threadIdx.x) >> 5);
  const int s0 = wave * 16;
  if (s0 >= Bn) return;            // uniform per wave
  const int smp = s0 + (lane & 15);

  // Per-sample features (duplicated in both lane halves on purpose).
  float xq[4];
  {
    const float4 xr = *(const float4*)(x + (size_t)smp * 4);
    xq[0] = xr.x; xq[1] = xr.y; xq[2] = xr.z; xq[3] = xr.w;
  }

  float Sr[8], Si[8];
  const int hb = (lane >> 4) * 8;
  #pragma unroll
  for (int r = 0; r < 8; ++r) {
    Sr[r] = ws[6912 + hb + r];
    Si[r] = ws[6928 + hb + r];
  }

  const v2f* __restrict__ wsA = (const v2f*)ws;

  for (int l = 0; l < N_LAYERS; ++l) {
    // ---------- per-sample data-encoding RX(lmbd*x) on each qubit ----------
    float cc[4], ssn[4];
    #pragma unroll
    for (int q = 0; q < 4; ++q) {
      const float a = 0.5f * lmbd[l*4 + q] * xq[q];
      cc[q]  = cosf(a);
      ssn[q] = sinf(a);
    }
    { // qubit 0 <-> amp bit 3: partner lives in the other lane half, same VGPR
      const float c = cc[0], s = ssn[0];
      #pragma unroll
      for (int r = 0; r < 8; ++r) {
        const float br = __shfl_xor(Sr[r], 16);
        const float bi = __shfl_xor(Si[r], 16);
        const float nr = c*Sr[r] + s*bi;   // RX: new = c*a - i*s*b
        const float ni = c*Si[r] - s*br;
        Sr[r] = nr; Si[r] = ni;
      }
    }
    #pragma unroll
    for (int q = 1; q < 4; ++q) {          // qubits 1..3: intra-lane, VGPR r ^ d
      const int d = 8 >> q;                // 4, 2, 1
      const float c = cc[q], s = ssn[q];
      float nr[8], ni[8];
      #pragma unroll
      for (int r = 0; r < 8; ++r) {
        nr[r] = c*Sr[r] + s*Si[r ^ d];
        ni[r] = c*Si[r] - s*Sr[r ^ d];
      }
      #pragma unroll
      for (int r = 0; r < 8; ++r) { Sr[r] = nr[r]; Si[r] = ni[r]; }
    }

    // ---------- shared-gate GEMM: S <- U_{l+1} * S via WMMA f32 16x16x4 ----------
    // Relayout D-layout state into B operands (K chunks of 4).
    v2f Br[4], Bi[4];
    #pragma unroll
    for (int c = 0; c < 4; ++c) {
      #pragma unroll
      for (int p = 0; p < 2; ++p) {
        const int k0 = 4*c + p;            // low half supplies K=k0, high half K=k0+2
        float br, bi;
        if (k0 < 8) {
          const float tr = __shfl_xor(Sr[k0 + 2], 16);
          const float ti = __shfl_xor(Si[k0 + 2], 16);
          br = (lane < 16) ? Sr[k0] : tr;
          bi = (lane < 16) ? Si[k0] : ti;
        } else {
          const float tr = __shfl_xor(Sr[k0 - 8], 16);
          const float ti = __shfl_xor(Si[k0 - 8], 16);
          br = (lane < 16) ? tr : Sr[k0 - 6];
          bi = (lane < 16) ? ti : Si[k0 - 6];
        }
        Br[c][p] = br; Bi[c][p] = bi;
      }
    }

    const int lu = l + 1;                  // apply U_1 .. U_8
    v2f Ar[4], Ai[4], An[4];
    #pragma unroll
    for (int c = 0; c < 4; ++c) {
      Ar[c] = wsA[((lu*3 + 0)*4 + c)*32 + lane];
      Ai[c] = wsA[((lu*3 + 1)*4 + c)*32 + lane];
      An[c] = wsA[((lu*3 + 2)*4 + c)*32 + lane];
    }

    v8f Dr = {};
    v8f Di = {};
    #pragma unroll
    for (int c = 0; c < 4; ++c)            // Dr += Re(U)*Re(S)
      Dr = __builtin_amdgcn_wmma_f32_16x16x4_f32(false, Ar[c], false, Br[c], (short)0, Dr, false, false);
    #pragma unroll
    for (int c = 0; c < 4; ++c)            // Dr += (-Im(U))*Im(S)
      Dr = __builtin_amdgcn_wmma_f32_16x16x4_f32(false, An[c], false, Bi[c], (short)0, Dr, false, false);
    #pragma unroll
    for (int c = 0; c < 4; ++c)            // Di += Im(U)*Re(S)
      Di = __builtin_amdgcn_wmma_f32_16x16x4_f32(false, Ai[c], false, Br[c], (short)0, Di, false, false);
    #pragma unroll
    for (int c = 0; c < 4; ++c)            // Di += Re(U)*Im(S)
      Di = __builtin_amdgcn_wmma_f32_16x16x4_f32(false, Ar[c], false, Bi[c], (short)0, Di, false, false);

    #pragma unroll
    for (int r = 0; r < 8; ++r) { Sr[r] = Dr[r]; Si[r] = Di[r]; }
  }

  // ---------- <Z0 Z1 Z2 Z3> = sum_m parity(m) * |amp_m|^2 ----------
  const float psgn[8] = { 1.f, -1.f, -1.f, 1.f, -1.f, 1.f, 1.f, -1.f };
  float acc = 0.f;
  #pragma unroll
  for (int r = 0; r < 8; ++r)
    acc += psgn[r] * (Sr[r]*Sr[r] + Si[r]*Si[r]);
  if (lane >= 16) acc = -acc;              // amps 8..15 have flipped parity
  acc += __shfl_xor(acc, 16);
  if (lane < 16) out[smp] = acc;
}

extern "C" void kernel_launch(void* const* d_in, const int* in_sizes, int n_in,
                              void* d_out, int out_size, void* d_ws, size_t ws_size,
                              hipStream_t stream)
{
  const float* x     = (const float*)d_in[0];   // [B,4] f32
  const float* theta = (const float*)d_in[1];   // [1,108] f32
  const float* lmbd  = (const float*)d_in[2];   // [32] f32
  float* out = (float*)d_out;                   // [B] f32
  float* ws  = (float*)d_ws;                    // needs 6944 floats (~28 KB)

  const int Bn = in_sizes[0] / N_QUBITS;

  pqc_setup<<<N_LAYERS + 1, 256, 0, stream>>>(theta, ws);

  const int waves   = (Bn + 15) / 16;           // one wave32 per 16 samples
  const int threads = 256;                      // 8 waves / block
  const int wpb     = threads / 32;
  const int blocks  = (waves + wpb - 1) / wpb;
  pqc_main<<<blocks, threads, 0, stream>>>(x, lmbd, ws, out, Bn);
}